// wmaeloss_85839216378484
// MI455X (gfx1250) — compile-verified
//
#include <hip/hip_runtime.h>

typedef __attribute__((ext_vector_type(2))) float v2f;
typedef __attribute__((ext_vector_type(8))) float v8f;

#define NBLOCKS  2048
#define NTHREADS 256
#define NWAVES   (NTHREADS / 32)

// Reduce one f32 value across the 32 lanes of a wave using
// V_WMMA_F32_16X16X4_F32 (full f32 precision).
//
// A (16x4, 2 VGPRs/lane): lane m (0-15) holds A[m][0],A[m][1]; lane m+16
// holds A[m][2],A[m][3]. We put the per-lane partial in BOTH A VGPRs.
// B (4x16, 2 VGPRs/lane): we set both VGPRs to 1.0 on lanes 0 and 16 only,
// which makes column 0 of B all-ones (robust to either row->VGPR striping),
// all other columns zero. Then D[m][0] = 2*(v_m + v_{m+16}).
// Lane 0 holds D[0..7][0] in its 8 D VGPRs, lane 16 holds D[8..15][0];
// one xor-16 fold + *0.5 yields the exact wave sum.
__device__ __forceinline__ float wave_reduce_wmma(float v) {
    const int lane = threadIdx.x & 31;
    v2f a; a.x = v; a.y = v;
    v2f b; b.x = 0.0f; b.y = 0.0f;
    if ((lane & 15) == 0) { b.x = 1.0f; b.y = 1.0f; }   // lanes 0 and 16
    v8f c = {};
    v8f d = __builtin_amdgcn_wmma_f32_16x16x4_f32(
        /*neg_a=*/false, a, /*neg_b=*/false, b,
        /*c_mod=*/(short)0, c, /*reuse_a=*/false, /*reuse_b=*/false);
    float s = d[0] + d[1] + d[2] + d[3] + d[4] + d[5] + d[6] + d[7];
    s += __shfl_xor(s, 16, 32);        // fold lanes 0 and 16 together
    s  = __shfl(s, 0, 32);             // broadcast lane 0 (others held zero columns)
    return s * 0.5f;
}

__device__ __forceinline__ void acc_elem(
    float yv, float pv,
    float e0, float e1, float e2, float e3, float e4, float e5, float e6, float e7,
    float w0, float w1, float w2, float w3, float w4, float w5, float w6,
    float& wd, float& cnt)
{
    float diff  = fabsf(pv - yv);
    bool  valid = (yv >= e0) && (yv < e7);
    // sorted edges -> select chain == searchsorted(side='right') - 1
    float w = yv < e1 ? w0 :
              yv < e2 ? w1 :
              yv < e3 ? w2 :
              yv < e4 ? w3 :
              yv < e5 ? w4 :
              yv < e6 ? w5 : w6;
    w   = valid ? w : 0.0f;
    wd  = fmaf(w, diff, wd);
    cnt += valid ? 1.0f : 0.0f;
}

__global__ __launch_bounds__(NTHREADS)
void wmae_partial_kernel(const float* __restrict__ y,
                         const float* __restrict__ p,
                         const float* __restrict__ weights,
                         const float* __restrict__ edge,
                         float* __restrict__ ws, int n)
{
    // Uniform tables -> scalar loads
    const float e0 = edge[0], e1 = edge[1], e2 = edge[2], e3 = edge[3];
    const float e4 = edge[4], e5 = edge[5], e6 = edge[6], e7 = edge[7];
    const float w0 = weights[0], w1 = weights[1], w2 = weights[2],
                w3 = weights[3], w4 = weights[4], w5 = weights[5], w6 = weights[6];

    float wd = 0.0f, cnt = 0.0f;

    const int nvec   = n >> 2;
    const int stride = gridDim.x * blockDim.x;
    const int gtid   = blockIdx.x * blockDim.x + threadIdx.x;
    const float4* __restrict__ y4 = (const float4*)y;
    const float4* __restrict__ p4 = (const float4*)p;

    for (int i = gtid; i < nvec; i += stride) {
        float4 yv = y4[i];
        float4 pv = p4[i];
        acc_elem(yv.x, pv.x, e0,e1,e2,e3,e4,e5,e6,e7, w0,w1,w2,w3,w4,w5,w6, wd, cnt);
        acc_elem(yv.y, pv.y, e0,e1,e2,e3,e4,e5,e6,e7, w0,w1,w2,w3,w4,w5,w6, wd, cnt);
        acc_elem(yv.z, pv.z, e0,e1,e2,e3,e4,e5,e6,e7, w0,w1,w2,w3,w4,w5,w6, wd, cnt);
        acc_elem(yv.w, pv.w, e0,e1,e2,e3,e4,e5,e6,e7, w0,w1,w2,w3,w4,w5,w6, wd, cnt);
    }
    // scalar tail (n not multiple of 4)
    const int base = nvec << 2;
    if (base + gtid < n) {
        acc_elem(y[base + gtid], p[base + gtid],
                 e0,e1,e2,e3,e4,e5,e6,e7, w0,w1,w2,w3,w4,w5,w6, wd, cnt);
    }

    // wave32 reduction via WMMA (EXEC is all-ones here: full 256-thread block,
    // uniform control flow)
    float wdw  = wave_reduce_wmma(wd);
    float cntw = wave_reduce_wmma(cnt);

    __shared__ float swd[NWAVES];
    __shared__ float scnt[NWAVES];
    const int lane = threadIdx.x & 31;
    const int wv   = threadIdx.x >> 5;
    if (lane == 0) { swd[wv] = wdw; scnt[wv] = cntw; }
    __syncthreads();
    if (threadIdx.x == 0) {
        float a = 0.0f, b = 0.0f;
        #pragma unroll
        for (int j = 0; j < NWAVES; ++j) { a += swd[j]; b += scnt[j]; }
        ws[2 * blockIdx.x + 0] = a;
        ws[2 * blockIdx.x + 1] = b;
    }
}

__global__ __launch_bounds__(NTHREADS)
void wmae_final_kernel(const float* __restrict__ ws,
                       float* __restrict__ out, int nblocks)
{
    float wd = 0.0f, cnt = 0.0f;
    for (int i = threadIdx.x; i < nblocks; i += NTHREADS) {
        wd  += ws[2 * i + 0];
        cnt += ws[2 * i + 1];
    }

    float wdw  = wave_reduce_wmma(wd);
    float cntw = wave_reduce_wmma(cnt);

    __shared__ float swd[NWAVES];
    __shared__ float scnt[NWAVES];
    const int lane = threadIdx.x & 31;
    const int wv   = threadIdx.x >> 5;
    if (lane == 0) { swd[wv] = wdw; scnt[wv] = cntw; }
    __syncthreads();
    if (threadIdx.x == 0) {
        float a = 0.0f, b = 0.0f;
        #pragma unroll
        for (int j = 0; j < NWAVES; ++j) { a += swd[j]; b += scnt[j]; }
        out[0] = a / b;
    }
}

extern "C" void kernel_launch(void* const* d_in, const int* in_sizes, int n_in,
                              void* d_out, int out_size, void* d_ws, size_t ws_size,
                              hipStream_t stream) {
    const float* y       = (const float*)d_in[0];
    const float* p       = (const float*)d_in[1];
    const float* weights = (const float*)d_in[2];
    const float* edge    = (const float*)d_in[3];
    float*       ws      = (float*)d_ws;
    float*       out     = (float*)d_out;
    const int    n       = in_sizes[0];

    wmae_partial_kernel<<<NBLOCKS, NTHREADS, 0, stream>>>(y, p, weights, edge, ws, n);
    wmae_final_kernel<<<1, NTHREADS, 0, stream>>>(ws, out, NBLOCKS);
}